// QuantumImageEncoder_6957847020215
// MI455X (gfx1250) — compile-verified
//
#include <hip/hip_runtime.h>
#include <math.h>

typedef __attribute__((ext_vector_type(2))) float v2f;
typedef __attribute__((ext_vector_type(4))) float v4f;
typedef __attribute__((ext_vector_type(8))) float v8f;

#define HW    50176
#define NQ    20
#define BATCH 16
#define CH    3
#define NTOT  (BATCH * HW)   // per-channel element count: 802816

// workspace layout (float offsets)
#define S_OFF   0            // raw strided sums  [48][20]      (960)
#define SSQ_OFF 960          // per-channel sum of squares [3]
#define MAX_OFF 963          // global max key (unsigned, order-preserving)
#define V_OFF   976          // v vectors  [16][20] x float4    (1280)
#define P_OFF   2304         // partial products [16][2][1024] x float2 (65536)

__device__ inline unsigned fkey(float f) {
    unsigned b = __float_as_uint(f);
    return (b & 0x80000000u) ? ~b : (b | 0x80000000u);
}
__device__ inline float funkey(unsigned k) {
    unsigned b = (k & 0x80000000u) ? (k ^ 0x80000000u) : ~k;
    return __uint_as_float(b);
}

__global__ void kInit(float* __restrict__ ws) {
    int t = blockIdx.x * blockDim.x + threadIdx.x;
    if (t < 964) ws[t] = 0.0f;   // S + sumsq + max key (0 == sentinel)
}

// One block per (b,c). Strided sums via V_WMMA_F32_16X16X4_F32:
//   A(m,k) = p[base + 16*s + m + 80*k]  (row m -> fixed q=(16s+m)%20, documented layout)
//   B      = ones(4x16)                  (layout-invariant)
//   D rows = row-sums of A (B==ones)     (documented C/D layout)
__global__ void __launch_bounds__(128)
kA_reduce(const float* __restrict__ x, float* __restrict__ ws)
{
    __shared__ float qbins[NQ];
    __shared__ float wsum[4][5][16];
    __shared__ float red[128];

    const int tid  = threadIdx.x;
    const int lane = tid & 31;
    const int wave = tid >> 5;
    const int bc   = blockIdx.x;                 // b*3 + c
    const float* __restrict__ p = x + (size_t)bc * HW;

    if (tid < NQ) qbins[tid] = 0.0f;
    __syncthreads();

    const int m    = lane & 15;                  // A row M = lane%16
    const int colA = (lane < 16) ? 0 : 2;        // VGPR0 holds K=0 (lanes 0-15) / K=2 (16-31)

    v2f ones; ones.x = 1.0f; ones.y = 1.0f;
    v8f acc[5];
    #pragma unroll
    for (int s = 0; s < 5; ++s) acc[s] = (v8f){0,0,0,0,0,0,0,0};

    float ssq  = 0.0f;
    float vmax = -3.402823466e38f;

    // 156 groups of 320 elements; 4 waves, 39 groups each (uniform -> EXEC all ones)
    for (int i = 0; i < 39; ++i) {
        const int base = (wave + 4 * i) * 320;
        #pragma unroll
        for (int s = 0; s < 5; ++s) {
            const float ax = p[base + 16 * s + m + 80 * colA];
            const float ay = p[base + 16 * s + m + 80 * (colA + 1)];
            ssq += ax * ax + ay * ay;
            vmax = fmaxf(vmax, fmaxf(ax, ay));
            v2f a; a.x = ax; a.y = ay;
            acc[s] = __builtin_amdgcn_wmma_f32_16x16x4_f32(
                false, a, false, ones, (short)0, acc[s], false, false);
        }
    }

    // D layout: lanes 0-15 hold rows M=c (component c), lanes 16-31 rows M=8+c.
    if (lane == 0 || lane == 16) {
        const int mo = (lane == 16) ? 8 : 0;
        #pragma unroll
        for (int s = 0; s < 5; ++s)
            #pragma unroll
            for (int c = 0; c < 8; ++c)
                wsum[wave][s][mo + c] = acc[s][c];
    }

    // tail: last 256 elements ((HW-256) % 20 == 0 -> q = t % 20)
    for (int t = tid; t < 256; t += 128) {
        const float v = p[HW - 256 + t];
        atomicAdd(&qbins[t % NQ], v);
        ssq += v * v;
        vmax = fmaxf(vmax, v);
    }

    red[tid] = ssq;
    __syncthreads();
    for (int off = 64; off > 0; off >>= 1) {
        if (tid < off) red[tid] += red[tid + off];
        __syncthreads();
    }
    if (tid == 0) atomicAdd(&ws[SSQ_OFF + (bc % CH)], red[0]);
    __syncthreads();
    red[tid] = vmax;
    __syncthreads();
    for (int off = 64; off > 0; off >>= 1) {
        if (tid < off) red[tid] = fmaxf(red[tid], red[tid + off]);
        __syncthreads();
    }
    if (tid == 0) atomicMax((unsigned*)&ws[MAX_OFF], fkey(red[0]));
    __syncthreads();

    if (tid < NQ) {
        float s = qbins[tid];
        for (int w = 0; w < 4; ++w) {
            #pragma unroll
            for (int k = 0; k < 5; ++k) {
                int mm = tid - 16 * k; mm %= NQ; if (mm < 0) mm += NQ;
                if (mm < 16) s += wsum[w][k][mm];   // q = (16k+m) % 20
            }
        }
        ws[S_OFF + bc * NQ + tid] = s;              // one block owns these 20 bins
    }
}

// One block, 320 threads: stats -> w -> gates -> per-(b,q) state vector v.
__global__ void __launch_bounds__(320)
kB_gates(const float* __restrict__ ep, float* __restrict__ ws)
{
    __shared__ float s_mean[CH], s_stdinv[CH], s_scale;
    const int tid = threadIdx.x;

    if (tid == 0) {
        const float mx = funkey(((const unsigned*)ws)[MAX_OFF]);
        s_scale = (mx > 1.0f) ? (1.0f / 255.0f) : 1.0f;
    }
    __syncthreads();
    if (tid < CH) {
        float tot = 0.0f;
        for (int b = 0; b < BATCH; ++b)
            for (int q = 0; q < NQ; ++q)
                tot += ws[S_OFF + (b * CH + tid) * NQ + q];
        const float scale = s_scale;
        const float mean  = scale * tot / (float)NTOT;
        const float ssq   = scale * scale * ws[SSQ_OFF + tid];
        float var = (ssq - (float)NTOT * mean * mean) / (float)(NTOT - 1);
        var = fmaxf(var, 0.0f);
        s_mean[tid]   = mean;
        s_stdinv[tid] = 1.0f / (sqrtf(var) + 1e-8f);
    }
    __syncthreads();

    const int b = tid / NQ, q = tid % NQ;
    const float cnt   = (q < 16) ? 2509.0f : 2508.0f;
    const float scale = s_scale;

    float v0r = 1.0f, v0i = 0.0f, v1r = 0.0f, v1i = 0.0f;
    for (int c = 0; c < CH; ++c) {
        const float w = (scale * ws[S_OFF + (b * CH + c) * NQ + q] / cnt - s_mean[c]) * s_stdinv[c];
        const float a  = 0.5f * w * ep[(c * NQ + q) * 3 + 0];
        const float bb = 0.5f * w * ep[(c * NQ + q) * 3 + 1];
        const float cc = 0.5f * w * ep[(c * NQ + q) * 3 + 2];
        float sa, ca, sb, cb, ec_i, ec_r;
        sincosf(a,  &sa,  &ca);
        sincosf(bb, &sb,  &cb);
        sincosf(cc, &ec_i, &ec_r);
        ec_i = -ec_i;                                  // e^{-i c}
        // M = Ry @ Rx
        const float M00r =  cb * ca, M00i =  sb * sa;
        const float M01r = -sb * ca, M01i = -cb * sa;
        const float M10r =  sb * ca, M10i = -cb * sa;
        const float M11r =  cb * ca, M11i = -sb * sa;
        // G = Rz @ M : row0 *= e^{-ic}, row1 *= e^{+ic}
        const float G00r = ec_r*M00r - ec_i*M00i, G00i = ec_r*M00i + ec_i*M00r;
        const float G01r = ec_r*M01r - ec_i*M01i, G01i = ec_r*M01i + ec_i*M01r;
        const float G10r = ec_r*M10r + ec_i*M10i, G10i = ec_r*M10i - ec_i*M10r;
        const float G11r = ec_r*M11r + ec_i*M11i, G11i = ec_r*M11i - ec_i*M11r;
        const float nv0r = G00r*v0r - G00i*v0i + G01r*v1r - G01i*v1i;
        const float nv0i = G00r*v0i + G00i*v0r + G01r*v1i + G01i*v1r;
        const float nv1r = G10r*v0r - G10i*v0i + G11r*v1r - G11i*v1i;
        const float nv1i = G10r*v0i + G10i*v0r + G11r*v1i + G11i*v1r;
        v0r = nv0r; v0i = nv0i; v1r = nv1r; v1i = nv1i;
    }
    v4f o; o.x = v0r; o.y = v0i; o.z = v1r; o.w = v1i;
    ((v4f*)(ws + V_OFF))[b * NQ + q] = o;
}

// Partial kron products over qubit halves: P[b][h][0..1023]
__global__ void __launch_bounds__(256)
kC_partial(float* __restrict__ ws)
{
    const int gt  = blockIdx.x * 256 + threadIdx.x;   // 0..32767
    const int b   = gt >> 11;
    const int h   = (gt >> 10) & 1;
    const int idx = gt & 1023;
    const v4f* __restrict__ vt = (const v4f*)(ws + V_OFF);
    float pr = 1.0f, pi = 0.0f;
    #pragma unroll
    for (int i = 0; i < 10; ++i) {
        const v4f v = vt[b * NQ + h * 10 + i];
        const int bit = (idx >> (9 - i)) & 1;
        const float fr = bit ? v.z : v.x;
        const float fi = bit ? v.w : v.y;
        const float nr = pr * fr - pi * fi;
        const float ni = pr * fi + pi * fr;
        pr = nr; pi = ni;
    }
    v2f o; o.x = pr; o.y = pi;
    ((v2f*)(ws + P_OFF))[(b * 2 + h) * 1024 + idx] = o;
}

// Outer product: s[b, j*1024+k] = P_hi[b][j] * P_lo[b][k]; 134 MB streamed out.
__global__ void __launch_bounds__(256)
kD_outer(const float* __restrict__ ws, float* __restrict__ out)
{
    const int blk = blockIdx.x;            // 0..16383
    const int b = blk >> 10;
    const int j = blk & 1023;
    const v2f* __restrict__ P = (const v2f*)(ws + P_OFF);
    const v2f ph = P[(b * 2 + 0) * 1024 + j];
    const int k0 = threadIdx.x * 4;
    const v4f* __restrict__ plo = (const v4f*)(P + (b * 2 + 1) * 1024 + k0); // 4 complex
    const v4f p01 = plo[0];
    const v4f p23 = plo[1];
    v4f o01, o23;
    o01.x = ph.x*p01.x - ph.y*p01.y;  o01.y = ph.x*p01.y + ph.y*p01.x;
    o01.z = ph.x*p01.z - ph.y*p01.w;  o01.w = ph.x*p01.w + ph.y*p01.z;
    o23.x = ph.x*p23.x - ph.y*p23.y;  o23.y = ph.x*p23.y + ph.y*p23.x;
    o23.z = ph.x*p23.z - ph.y*p23.w;  o23.w = ph.x*p23.w + ph.y*p23.z;
    v4f* dst = (v4f*)(out + ((((size_t)b << 20) | (size_t)(j << 10) | (size_t)k0) << 1));
    __builtin_nontemporal_store(o01, dst);
    __builtin_nontemporal_store(o23, dst + 1);
}

extern "C" void kernel_launch(void* const* d_in, const int* in_sizes, int n_in,
                              void* d_out, int out_size, void* d_ws, size_t ws_size,
                              hipStream_t stream)
{
    (void)in_sizes; (void)n_in; (void)out_size; (void)ws_size;
    const float* x  = (const float*)d_in[0];   // [16,3,224,224] f32
    const float* ep = (const float*)d_in[1];   // [3,20,3] f32
    float* ws  = (float*)d_ws;
    float* out = (float*)d_out;                // [16, 2^20] complex64 as f32 pairs

    hipLaunchKernelGGL(kInit,      dim3(4),     dim3(256), 0, stream, ws);
    hipLaunchKernelGGL(kA_reduce,  dim3(48),    dim3(128), 0, stream, x, ws);
    hipLaunchKernelGGL(kB_gates,   dim3(1),     dim3(320), 0, stream, ep, ws);
    hipLaunchKernelGGL(kC_partial, dim3(128),   dim3(256), 0, stream, ws);
    hipLaunchKernelGGL(kD_outer,   dim3(16384), dim3(256), 0, stream, ws, out);
}